// DarkChannelPriorLayerV2_9371618640650
// MI455X (gfx1250) — compile-verified
//
#include <hip/hip_runtime.h>

// ---------------------------------------------------------------------------
// Dark-channel prior: 15x15 sliding-window min, zero-padded (SAME), NHWC f32.
// Separable min (row pass then column pass) staged through LDS.
// CDNA5 path: global_load_async_to_lds_b32 + s_wait_asynccnt (ASYNCcnt).
// ---------------------------------------------------------------------------

#define K     15
#define RAD   7
#define TILE  32
#define IN_T  (TILE + K - 1)      // 46 rows/cols of input halo tile
#define HH    512
#define WW    512
#define CC    3
#define ROWF  (IN_T * CC)         // 138 floats per staged input row
#define ROWP  144                 // padded LDS row stride (floats)
#define OUTF  (TILE * CC)         // 96 floats per output row
#define NTHREADS 256

#if __has_builtin(__builtin_amdgcn_global_load_async_to_lds_b32)
#define HAVE_ASYNC_LDS 1
#endif

// Builtin signature (from hipcc diagnostic): param 0 is AS(1) int*, so cast
// both pointers to non-const int* in their respective address spaces.
typedef __attribute__((address_space(1))) int* gint_p;
typedef __attribute__((address_space(3))) int* lint_p;

__global__ __launch_bounds__(NTHREADS) void dark_channel_min15(
    const float* __restrict__ in, float* __restrict__ out) {
  __shared__ float tin[IN_T][ROWP];   // staged input tile (+halo), zero-padded
  __shared__ float hmn[IN_T][OUTF];   // horizontal (row) min intermediate

  const int t  = threadIdx.x;
  const int bx = blockIdx.x;
  const int by = blockIdx.y;
  const int b  = blockIdx.z;
  const int x0 = bx * TILE;
  const int y0 = by * TILE;

  // Only tiles whose halo leaves the image need the zero pre-fill (block-uniform).
  const bool border = (bx == 0) | (by == 0) |
                      (bx == (int)gridDim.x - 1) | (by == (int)gridDim.y - 1);
  if (border) {
    float* flat = &tin[0][0];
    for (int i = t; i < IN_T * ROWP; i += NTHREADS) flat[i] = 0.0f;
  }
  __syncthreads();

  const long batch_base = (long)b * HH * WW * CC;

  // ---- Stage input tile into LDS with gfx1250 async copies -----------------
  // Element i of the tile: row r (0..45), k (0..137) runs over 46 cols x 3 ch,
  // which is a contiguous 138-float span of global memory per row.
  for (int i = t; i < IN_T * ROWF; i += NTHREADS) {
    const int r   = i / ROWF;
    const int k   = i - r * ROWF;
    const int gy  = y0 - RAD + r;
    const int col = k / CC;
    const int gx  = x0 - RAD + col;
    if (gy >= 0 && gy < HH && gx >= 0 && gx < WW) {
      const long gidx = batch_base + ((long)gy * WW + (x0 - RAD)) * CC + k;
#ifdef HAVE_ASYNC_LDS
      __builtin_amdgcn_global_load_async_to_lds_b32(
          (gint_p)(in + gidx), (lint_p)&tin[r][k], 0, 0);
#else
      tin[r][k] = in[gidx];
#endif
    }
  }

#ifdef HAVE_ASYNC_LDS
#if __has_builtin(__builtin_amdgcn_s_wait_asynccnt)
  __builtin_amdgcn_s_wait_asynccnt(0);
#else
  asm volatile("s_wait_asynccnt 0x0" ::: "memory");
#endif
#endif
  __syncthreads();

  // ---- Horizontal pass: 15-wide row min ------------------------------------
  // hmn[r][x*3+c] = min_{dx=0..14} tin[r][(x+dx)*3 + c]
  for (int j = t; j < IN_T * OUTF; j += NTHREADS) {
    const int r = j / OUTF;
    const int k = j - r * OUTF;
    float m = tin[r][k];
#pragma unroll
    for (int dx = 1; dx < K; ++dx) m = fminf(m, tin[r][k + CC * dx]);
    hmn[r][k] = m;
  }
  __syncthreads();

  // ---- Vertical pass: 15-tall column min + coalesced store -----------------
  for (int j = t; j < TILE * OUTF; j += NTHREADS) {
    const int y = j / OUTF;
    const int k = j - y * OUTF;
    float m = hmn[y][k];
#pragma unroll
    for (int dy = 1; dy < K; ++dy) m = fminf(m, hmn[y + dy][k]);
    const int gy = y0 + y;
    out[batch_base + ((long)gy * WW + x0) * CC + k] = m;
  }
}

extern "C" void kernel_launch(void* const* d_in, const int* in_sizes, int n_in,
                              void* d_out, int out_size, void* d_ws, size_t ws_size,
                              hipStream_t stream) {
  (void)n_in; (void)out_size; (void)d_ws; (void)ws_size;
  const float* in = (const float*)d_in[0];
  float* out      = (float*)d_out;
  const int batches = in_sizes[0] / (HH * WW * CC);   // 32 for the reference
  dim3 grid(WW / TILE, HH / TILE, batches);
  dark_channel_min15<<<grid, NTHREADS, 0, stream>>>(in, out);
}